// OptNetDenoiser_61435212202786
// MI455X (gfx1250) — compile-verified
//
#include <hip/hip_runtime.h>

typedef __attribute__((ext_vector_type(2))) float v2f;
typedef __attribute__((ext_vector_type(8))) float v8f;

namespace {

constexpr int   Tn   = 256;   // x dimension
constexpr int   M1   = 255;   // rows of D (half the constraint count)
constexpr int   MS   = 257;   // padded LDS row stride (257 % 64 == 1 -> bank-conflict-free columns)
constexpr int   NIT  = 30;
constexpr float SIG  = 0.1f;

// ---- shared-memory layout (float offsets) ----
constexpr int OFF_M   = 0;
constexpr int OFF_X   = OFF_M   + 256 * MS;   // x        [256]
constexpr int OFF_RHS = OFF_X   + 256;        // rhs / dx [256]
constexpr int OFF_LA  = OFF_RHS + 256;        // lambda1  [256]
constexpr int OFF_LB  = OFF_LA  + 256;        // lambda2  [256]
constexpr int OFF_SA  = OFF_LB  + 256;        // s1       [256]
constexpr int OFF_SB  = OFF_SA  + 256;        // s2       [256]
constexpr int OFF_W   = OFF_SB  + 256;        // w1+w2    [256] (ws[255]=0)
constexpr int OFF_V   = OFF_W   + 256;        // rhs-matvec vector [256]
constexpr int OFF_GX  = OFF_V   + 256;        // D*x      [256]
constexpr int OFF_GDX = OFF_GX  + 256;        // D*dx     [256]
constexpr int OFF_RED = OFF_GDX + 256;        // reduction scratch [8]
constexpr int SMEM_F  = OFF_RED + 8;
constexpr size_t SMEM_BYTES = (size_t)SMEM_F * sizeof(float);
static_assert(SMEM_BYTES < 320u * 1024u, "fits CDNA5 320KB LDS per WGP");

__device__ inline float block_sum(float v, float* red, int tid) {
  const int lane = tid & 31, w = tid >> 5;
  #pragma unroll
  for (int o = 16; o; o >>= 1) v += __shfl_xor(v, o, 32);
  if (lane == 0) red[w] = v;
  __syncthreads();
  float r = red[0];
  #pragma unroll
  for (int i = 1; i < 8; ++i) r += red[i];
  __syncthreads();
  return r;
}

__device__ inline float block_min(float v, float* red, int tid) {
  const int lane = tid & 31, w = tid >> 5;
  #pragma unroll
  for (int o = 16; o; o >>= 1) v = fminf(v, __shfl_xor(v, o, 32));
  if (lane == 0) red[w] = v;
  __syncthreads();
  float r = red[0];
  #pragma unroll
  for (int i = 1; i < 8; ++i) r = fminf(r, red[i]);
  __syncthreads();
  return r;
}

// One workgroup (256 threads = 8 wave32) solves one batch element's QP:
//   min 0.5 x^T x - y^T x   s.t. [D;-D] x <= h*1
// All IPM state + the 256x256 reduced-KKT matrix live in LDS.
// WMMA sites: (1) M = I + D^T diag(w) D formation, (2) blocked-Cholesky
// trailing updates C -= P_i P_j^T (both on V_WMMA_F32_16X16X4_F32).
__global__ __launch_bounds__(256, 1)
void ipm_qp_kernel(const float* __restrict__ y, const float* __restrict__ D,
                   const float* __restrict__ theta, float* __restrict__ out)
{
  extern __shared__ float sm[];
  const int tid  = threadIdx.x;
  const int lane = tid & 31;
  const int wid  = tid >> 5;
  const int half = lane >> 4;   // K sub-block select per ISA A/B layout
  const int li   = lane & 15;
  const int b    = blockIdx.x;
  const float h  = theta[0];
  const float* __restrict__ yb = y + (size_t)b * Tn;

  float* Ms  = sm + OFF_M;
  float* xs  = sm + OFF_X;
  float* rhs = sm + OFF_RHS;
  float* la  = sm + OFF_LA;
  float* lb  = sm + OFF_LB;
  float* sa  = sm + OFF_SA;
  float* sb  = sm + OFF_SB;
  float* ws  = sm + OFF_W;
  float* vv  = sm + OFF_V;
  float* gx  = sm + OFF_GX;
  float* gdx = sm + OFF_GDX;
  float* red = sm + OFF_RED;

  // warm L2 with D (255*256 floats); emits global_prefetch_b8
  for (size_t off = (size_t)tid * 64; off < (size_t)M1 * Tn; off += 256u * 64u)
    __builtin_prefetch(&D[off], 0, 1);

  xs[tid] = 0.f; la[tid] = 1.f; lb[tid] = 1.f; sa[tid] = 1.f; sb[tid] = 1.f;
  __syncthreads();

  for (int it = 0; it < NIT; ++it) {
    // ---- gx = D x : row matvec, 8 waves x coalesced lane-strided reads ----
    for (int k = wid; k < M1; k += 8) {
      float acc = 0.f;
      for (int t0 = lane; t0 < Tn; t0 += 32) acc += D[k * Tn + t0] * xs[t0];
      #pragma unroll
      for (int o = 16; o; o >>= 1) acc += __shfl_xor(acc, o, 32);
      if (lane == 0) gx[k] = acc;
    }
    __syncthreads();

    // ---- mu = mean(lam * s) over all 2*M1 constraints ----
    const float prod = (tid < M1) ? (la[tid] * sa[tid] + lb[tid] * sb[tid]) : 0.f;
    const float mu = block_sum(prod, red, tid) * (1.f / (float)(2 * M1));

    // ---- per-constraint quantities (kept in registers across the solve) ----
    float lA = 0.f, lB = 0.f, sA = 0.f, sB = 0.f;
    float rp1 = 0.f, rp2 = 0.f, rc1 = 0.f, rc2 = 0.f;
    if (tid < M1) {
      lA = la[tid]; lB = lb[tid]; sA = sa[tid]; sB = sb[tid];
      const float g = gx[tid];
      rp1 =  g + sA - h;                 // r_p upper half (D x + s1 - h)
      rp2 = -g + sB - h;                 // r_p lower half (-D x + s2 - h)
      rc1 = lA * sA - SIG * mu;
      rc2 = lB * sB - SIG * mu;
      const float u1 = (rc1 - lA * rp1) / sA;
      const float u2 = (rc2 - lB * rp2) / sB;
      vv[tid] = (u1 - u2) - (lA - lB);   // rhs = (y-x) + D^T vv
      ws[tid] = lA / sA + lB / sB;       // diag weight for M
    } else { vv[tid] = 0.f; ws[tid] = 0.f; }   // ws[255]=0 kills K padding
    __syncthreads();

    // ---- rhs = (y - x) + D^T vv : column matvec, coalesced across threads ----
    {
      float acc = yb[tid] - xs[tid];
      for (int k = 0; k < M1; ++k) acc += D[k * Tn + tid] * vv[k];
      rhs[tid] = acc;
    }

    // ---- M = I + D^T diag(ws) D, lower triangle, f32 WMMA 16x16x4 ----
    // 136 lower-triangular 16x16 tiles = exactly 17 per wave (no divergence,
    // EXEC all-ones at every v_wmma as the ISA requires).
    for (int t = wid; t < 136; t += 8) {
      int ti = 0;
      while (((ti + 1) * (ti + 2)) / 2 <= t) ++ti;
      const int tj = t - (ti * (ti + 1)) / 2;
      const int I0 = ti * 16, J0 = tj * 16;
      v8f c = {};
      for (int k0 = 0; k0 < 256; k0 += 4) {
        const int ka  = k0 + 2 * half;                   // <= 254
        const int ka1 = (ka + 1 < M1) ? ka + 1 : M1 - 1; // clamp (branch-free)
        const float w0 = ws[ka];
        const float w1 = ws[ka + 1];                     // ws[255]==0 -> padded row contributes 0
        v2f A, B;
        A.x = D[ka  * Tn + I0 + li] * w0;                // A[M][K] = w*D^T tile
        A.y = D[ka1 * Tn + I0 + li] * w1;
        B.x = D[ka  * Tn + J0 + li];                     // B[K][N] = D tile
        B.y = D[ka1 * Tn + J0 + li];
        c = __builtin_amdgcn_wmma_f32_16x16x4_f32(false, A, false, B,
                                                  (short)0, c, false, false);
      }
      #pragma unroll
      for (int r = 0; r < 8; ++r) {                      // C/D layout: M=r+8*half, N=li
        const int row = I0 + r + 8 * half;
        const int col = J0 + li;
        Ms[row * MS + col] = c[r] + (row == col ? 1.f : 0.f);
      }
    }
    __syncthreads();

    // ==== blocked Cholesky, NB = 16, trailing updates on WMMA ====
    for (int kb = 0; kb < 16; ++kb) {
      const int K0 = kb * 16;

      // (1) factor 16x16 diagonal block in-place (thread 0, left-looking)
      if (tid == 0) {
        for (int c = 0; c < 16; ++c) {
          float* rowc = &Ms[(K0 + c) * MS + K0];
          float dv = rowc[c];
          for (int d = 0; d < c; ++d) dv -= rowc[d] * rowc[d];
          dv = sqrtf(dv);
          rowc[c] = dv;
          for (int r = c + 1; r < 16; ++r) {
            float* rowr = &Ms[(K0 + r) * MS + K0];
            float v = rowr[c];
            for (int d = 0; d < c; ++d) v -= rowr[d] * rowc[d];
            rowr[c] = v / dv;
          }
        }
      }
      __syncthreads();

      // (2) panel triangular solve: one row per thread, diag block broadcast
      {
        const int i = K0 + 16 + tid;
        if (i < Tn) {
          float* rowi = &Ms[i * MS + K0];
          #pragma unroll
          for (int c = 0; c < 16; ++c) {
            const float* rowc = &Ms[(K0 + c) * MS + K0];
            float v = rowi[c];
            for (int d = 0; d < c; ++d) v -= rowi[d] * rowc[d];
            rowi[c] = v / rowc[c];
          }
        }
      }
      __syncthreads();

      // (3) trailing update C -= P_i P_j^T : 16x16x16 GEMMs on WMMA from LDS
      //     (f32 WMMA has no A/B-neg modifier -> negate A at load)
      {
        const int nrem   = 15 - kb;
        const int ntiles = (nrem * (nrem + 1)) / 2;
        for (int t = wid; t < ntiles; t += 8) {
          int ti2 = 0;
          while (((ti2 + 1) * (ti2 + 2)) / 2 <= t) ++ti2;
          const int tj2 = t - (ti2 * (ti2 + 1)) / 2;
          const int I0 = (kb + 1 + ti2) * 16;
          const int J0 = (kb + 1 + tj2) * 16;
          v8f c;
          #pragma unroll
          for (int r = 0; r < 8; ++r)
            c[r] = Ms[(I0 + r + 8 * half) * MS + J0 + li];
          #pragma unroll
          for (int k0 = 0; k0 < 16; k0 += 4) {
            const int kc = K0 + k0 + 2 * half;
            v2f A, B;
            A.x = -Ms[(I0 + li) * MS + kc];       // A[M][K] = -L panel (row I0+li)
            A.y = -Ms[(I0 + li) * MS + kc + 1];
            B.x =  Ms[(J0 + li) * MS + kc];       // B[K][N] = L panel^T (row J0+li)
            B.y =  Ms[(J0 + li) * MS + kc + 1];
            c = __builtin_amdgcn_wmma_f32_16x16x4_f32(false, A, false, B,
                                                      (short)0, c, false, false);
          }
          #pragma unroll
          for (int r = 0; r < 8; ++r)
            Ms[(I0 + r + 8 * half) * MS + J0 + li] = c[r];
        }
      }
      __syncthreads();
    }

    // ==== blocked forward solve L v = rhs ====
    for (int kb = 0; kb < 16; ++kb) {
      const int K0 = kb * 16;
      if (tid == 0) {
        for (int c = 0; c < 16; ++c) {
          const float* rowc = &Ms[(K0 + c) * MS + K0];
          float v = rhs[K0 + c];
          for (int d = 0; d < c; ++d) v -= rowc[d] * rhs[K0 + d];
          rhs[K0 + c] = v / rowc[c];
        }
      }
      __syncthreads();
      {
        const int i = K0 + 16 + tid;
        if (i < Tn) {
          const float* rowi = &Ms[i * MS + K0];
          float acc = 0.f;
          #pragma unroll
          for (int c = 0; c < 16; ++c) acc += rowi[c] * rhs[K0 + c];
          rhs[i] -= acc;
        }
      }
      __syncthreads();
    }
    // ==== blocked backward solve L^T dx = v   (L^T[i][j] = Ms[j][i]) ====
    for (int kb = 15; kb >= 0; --kb) {
      const int K0 = kb * 16;
      if (tid == 0) {
        for (int c = 15; c >= 0; --c) {
          float v = rhs[K0 + c];
          for (int d = c + 1; d < 16; ++d) v -= Ms[(K0 + d) * MS + K0 + c] * rhs[K0 + d];
          rhs[K0 + c] = v / Ms[(K0 + c) * MS + K0 + c];
        }
      }
      __syncthreads();
      if (tid < K0) {
        float acc = 0.f;
        #pragma unroll
        for (int c = 0; c < 16; ++c) acc += Ms[(K0 + c) * MS + tid] * rhs[K0 + c];
        rhs[tid] -= acc;
      }
      __syncthreads();
    }
    // rhs now holds dx

    // ---- gdx = D dx ----
    for (int k = wid; k < M1; k += 8) {
      float acc = 0.f;
      for (int t0 = lane; t0 < Tn; t0 += 32) acc += D[k * Tn + t0] * rhs[t0];
      #pragma unroll
      for (int o = 16; o; o >>= 1) acc += __shfl_xor(acc, o, 32);
      if (lane == 0) gdx[k] = acc;
    }
    __syncthreads();

    // ---- ds, dlam, fraction-to-boundary step length ----
    float rmin = 3.4e38f;
    float d1 = 0.f, d2 = 0.f, e1 = 0.f, e2 = 0.f;
    if (tid < M1) {
      const float gd = gdx[tid];
      d1 = -rp1 - gd;
      d2 = -rp2 + gd;
      e1 = (-rc1 - lA * d1) / sA;
      e2 = (-rc2 - lB * d2) / sB;
      if (d1 < 0.f) rmin = fminf(rmin, -sA / d1);
      if (d2 < 0.f) rmin = fminf(rmin, -sB / d2);
      if (e1 < 0.f) rmin = fminf(rmin, -lA / e1);
      if (e2 < 0.f) rmin = fminf(rmin, -lB / e2);
    }
    const float a = fminf(1.f, 0.99f * block_min(rmin, red, tid));

    // ---- updates ----
    xs[tid] += a * rhs[tid];
    if (tid < M1) {
      la[tid] += a * e1;
      lb[tid] += a * e2;
      sa[tid] += a * d1;
      sb[tid] += a * d2;
    }
    __syncthreads();
  }

  out[(size_t)b * Tn + tid] = xs[tid];
}

} // namespace

extern "C" void kernel_launch(void* const* d_in, const int* in_sizes, int n_in,
                              void* d_out, int out_size, void* d_ws, size_t ws_size,
                              hipStream_t stream) {
  (void)n_in; (void)out_size; (void)d_ws; (void)ws_size;
  const float* y     = (const float*)d_in[0];   // [B, 256] f32
  const float* D     = (const float*)d_in[1];   // [255, 256] f32
  const float* theta = (const float*)d_in[2];   // [1] f32
  float* out = (float*)d_out;                   // [B, 256] f32
  const int B = in_sizes[0] / Tn;               // 512

  hipFuncSetAttribute(reinterpret_cast<const void*>(ipm_qp_kernel),
                      hipFuncAttributeMaxDynamicSharedMemorySize,
                      (int)SMEM_BYTES);
  ipm_qp_kernel<<<dim3(B), dim3(256), SMEM_BYTES, stream>>>(y, D, theta, out);
}